// GAT_17428977287559
// MI455X (gfx1250) — compile-verified
//
#include <hip/hip_runtime.h>
#include <math.h>

typedef __attribute__((ext_vector_type(2))) float v2f;
typedef __attribute__((ext_vector_type(8))) float v8f;

#define LRELU_SLOPE 0.2f

// ---------------------------------------------------------------------------
// Dense projection: C[nstrips*16, FOUT] = A[nstrips*16, FIN] @ W[FIN, FOUT]
// One wave per 16-row strip, all FOUT columns held in v8f accumulators.
// Uses V_WMMA_F32_16X16X4_F32 (full-f32 matrix op, f32 accumulate).
// A-frag (16x4 f32): lanes 0-15 -> K=0,1 ; lanes 16-31 -> K=2,3 (ISA 7.12.2)
// B-frag (4x16 f32): lane = N   ; same K split across lane halves
// C/D  : VGPR r holds M=r (lanes 0-15) / M=r+8 (lanes 16-31), N=lane&15
// ---------------------------------------------------------------------------
template <int FIN, int FOUT>
__global__ __launch_bounds__(256) void gemm_wmma_kernel(
    const float* __restrict__ A, const float* __restrict__ W,
    float* __restrict__ C, int nstrips) {
  const int wave = (int)((blockIdx.x * blockDim.x + threadIdx.x) >> 5);
  if (wave >= nstrips) return;  // wave-uniform: EXEC stays all-ones for WMMA
  const int lane = threadIdx.x & 31;
  const int half = lane >> 4;
  const int m = lane & 15;
  constexpr int JT = FOUT / 16;

  const float* __restrict__ arow = A + (size_t)(wave * 16 + m) * FIN + 2 * half;

  v8f acc[JT];
  const v8f vzero = {0.f, 0.f, 0.f, 0.f, 0.f, 0.f, 0.f, 0.f};
#pragma unroll
  for (int j = 0; j < JT; ++j) acc[j] = vzero;

  for (int k = 0; k < FIN; k += 4) {
    v2f a;
    a.x = arow[k];
    a.y = arow[k + 1];
    const float* __restrict__ wrow = W + (size_t)(k + 2 * half) * FOUT + m;
#pragma unroll
    for (int j = 0; j < JT; ++j) {
      v2f b;
      b.x = wrow[j * 16];
      b.y = wrow[j * 16 + FOUT];
      acc[j] = __builtin_amdgcn_wmma_f32_16x16x4_f32(
          false, a, false, b, (short)0, acc[j], false, false);
    }
  }

  const int rbase = wave * 16 + 8 * half;
#pragma unroll
  for (int j = 0; j < JT; ++j) {
#pragma unroll
    for (int r = 0; r < 8; ++r) {
      C[(size_t)(rbase + r) * FOUT + j * 16 + m] = acc[j][r];
    }
  }
}

// ---------------------------------------------------------------------------
// el[n,h] = sum_d feat[n,h,d]*al[h,d] ; er likewise. One thread per (n,h).
// ---------------------------------------------------------------------------
template <int H, int D>
__global__ void attn_scores_kernel(const float* __restrict__ feat,
                                   const float* __restrict__ al,
                                   const float* __restrict__ ar,
                                   float* __restrict__ el,
                                   float* __restrict__ er, int n) {
  int tid = blockIdx.x * blockDim.x + threadIdx.x;
  if (tid >= n * H) return;
  const int node = tid / H;
  const int h = tid % H;
  const float* __restrict__ f = feat + (size_t)node * (H * D) + h * D;
  float sl = 0.f, sr = 0.f;
#pragma unroll
  for (int d = 0; d < D; ++d) {
    const float v = f[d];
    sl += v * al[h * D + d];
    sr += v * ar[h * D + d];
  }
  el[tid] = sl;
  er[tid] = sr;
}

__global__ void fill_kernel(float* __restrict__ p, float v, int n) {
  int tid = blockIdx.x * blockDim.x + threadIdx.x;
  if (tid < n) p[tid] = v;
}

__device__ __forceinline__ float leaky(float x) {
  return x > 0.f ? x : LRELU_SLOPE * x;
}

// pass 1: segment max over incoming edges (global_atomic_max_num_f32)
template <int H>
__global__ void edge_max_kernel(const int* __restrict__ src,
                                const int* __restrict__ dst,
                                const float* __restrict__ el,
                                const float* __restrict__ er,
                                float* __restrict__ emax, int E) {
  int e = blockIdx.x * blockDim.x + threadIdx.x;
  if (e >= E) return;
  const int s = src[e], d = dst[e];
#pragma unroll
  for (int h = 0; h < H; ++h) {
    const float x = leaky(el[s * H + h] + er[d * H + h]);
    atomicMax(&emax[d * H + h], x);
  }
}

// pass 2: softmax denominator (global_atomic_add_f32)
template <int H>
__global__ void edge_denom_kernel(const int* __restrict__ src,
                                  const int* __restrict__ dst,
                                  const float* __restrict__ el,
                                  const float* __restrict__ er,
                                  const float* __restrict__ emax,
                                  float* __restrict__ denom, int E) {
  int e = blockIdx.x * blockDim.x + threadIdx.x;
  if (e >= E) return;
  const int s = src[e], d = dst[e];
#pragma unroll
  for (int h = 0; h < H; ++h) {
    const float x = leaky(el[s * H + h] + er[d * H + h]);
    atomicAdd(&denom[d * H + h], expf(x - emax[d * H + h]));
  }
}

// pass 3: rst[dst] += alpha * feat[src]. One thread per (edge, head).
template <int H, int D>
__global__ void edge_scatter_kernel(const int* __restrict__ src,
                                    const int* __restrict__ dst,
                                    const float* __restrict__ el,
                                    const float* __restrict__ er,
                                    const float* __restrict__ emax,
                                    const float* __restrict__ denom,
                                    const float* __restrict__ feat,
                                    float* __restrict__ rst, int E) {
  int tid = blockIdx.x * blockDim.x + threadIdx.x;
  if (tid >= E * H) return;
  const int e = tid / H;
  const int h = tid % H;
  const int s = src[e], d = dst[e];
  const float x = leaky(el[s * H + h] + er[d * H + h]);
  const float dn = denom[d * H + h];
  const float alpha = expf(x - emax[d * H + h]) / (dn == 0.f ? 1.f : dn);
  const float* __restrict__ fp = feat + (size_t)s * (H * D) + h * D;
  float* __restrict__ rp = rst + (size_t)d * (H * D) + h * D;
#pragma unroll
  for (int i = 0; i < D; ++i) atomicAdd(&rp[i], fp[i] * alpha);
}

// finalize: += bias, optional ELU (alpha=1)
template <bool ELU>
__global__ void finalize_kernel(float* __restrict__ rst,
                                const float* __restrict__ bias, int n, int HD) {
  int tid = blockIdx.x * blockDim.x + threadIdx.x;
  if (tid >= n * HD) return;
  float x = rst[tid] + bias[tid % HD];
  if (ELU) x = x > 0.f ? x : expm1f(x);
  rst[tid] = x;
}

static inline int cdiv(long a, long b) { return (int)((a + b - 1) / b); }

extern "C" void kernel_launch(void* const* d_in, const int* in_sizes, int n_in,
                              void* d_out, int out_size, void* d_ws,
                              size_t ws_size, hipStream_t stream) {
  const float* features = (const float*)d_in[0];
  const int* esrc = (const int*)d_in[1];
  const int* edst = (const int*)d_in[2];
  const float* W0 = (const float*)d_in[3];
  const float* al0 = (const float*)d_in[4];
  const float* ar0 = (const float*)d_in[5];
  const float* b0 = (const float*)d_in[6];
  const float* W1 = (const float*)d_in[7];
  const float* al1 = (const float*)d_in[8];
  const float* ar1 = (const float*)d_in[9];
  const float* b1 = (const float*)d_in[10];
  const float* W2 = (const float*)d_in[11];
  const float* al2 = (const float*)d_in[12];
  const float* ar2 = (const float*)d_in[13];
  const float* b2 = (const float*)d_in[14];

  const int N = in_sizes[0] / 256;  // 50000
  const int E = in_sizes[1];        // 800000
  const int NSTRIPS = N / 16;       // 50000 = 16 * 3125

  // workspace layout (floats), all L2-resident (~58 MB total)
  float* ws = (float*)d_ws;
  float* bufA = ws;                    // [N,128] GEMM output (feat)
  float* bufB = bufA + (size_t)N * 128;  // [N,128] aggregation output / next input
  float* el = bufB + (size_t)N * 128;  // [N,8]
  float* er = el + (size_t)N * 8;      // [N,8]
  float* emax = er + (size_t)N * 8;    // [N,8]
  float* denom = emax + (size_t)N * 8; // [N,8]
  float* outp = (float*)d_out;         // [N,64]

  const int B = 256;
  const float NEG_INF = -__builtin_huge_valf();

  // ------------------------- Layer 0: 256 -> 8x16 -------------------------
  gemm_wmma_kernel<256, 128>
      <<<cdiv((long)NSTRIPS * 32, B), B, 0, stream>>>(features, W0, bufA, NSTRIPS);
  attn_scores_kernel<8, 16>
      <<<cdiv((long)N * 8, B), B, 0, stream>>>(bufA, al0, ar0, el, er, N);
  fill_kernel<<<cdiv((long)N * 8, B), B, 0, stream>>>(emax, NEG_INF, N * 8);
  hipMemsetAsync(denom, 0, (size_t)N * 8 * sizeof(float), stream);
  hipMemsetAsync(bufB, 0, (size_t)N * 128 * sizeof(float), stream);
  edge_max_kernel<8><<<cdiv(E, B), B, 0, stream>>>(esrc, edst, el, er, emax, E);
  edge_denom_kernel<8>
      <<<cdiv(E, B), B, 0, stream>>>(esrc, edst, el, er, emax, denom, E);
  edge_scatter_kernel<8, 16><<<cdiv((long)E * 8, B), B, 0, stream>>>(
      esrc, edst, el, er, emax, denom, bufA, bufB, E);
  finalize_kernel<true>
      <<<cdiv((long)N * 128, B), B, 0, stream>>>(bufB, b0, N, 128);

  // ------------------------- Layer 1: 128 -> 8x16 -------------------------
  gemm_wmma_kernel<128, 128>
      <<<cdiv((long)NSTRIPS * 32, B), B, 0, stream>>>(bufB, W1, bufA, NSTRIPS);
  attn_scores_kernel<8, 16>
      <<<cdiv((long)N * 8, B), B, 0, stream>>>(bufA, al1, ar1, el, er, N);
  fill_kernel<<<cdiv((long)N * 8, B), B, 0, stream>>>(emax, NEG_INF, N * 8);
  hipMemsetAsync(denom, 0, (size_t)N * 8 * sizeof(float), stream);
  hipMemsetAsync(bufB, 0, (size_t)N * 128 * sizeof(float), stream);
  edge_max_kernel<8><<<cdiv(E, B), B, 0, stream>>>(esrc, edst, el, er, emax, E);
  edge_denom_kernel<8>
      <<<cdiv(E, B), B, 0, stream>>>(esrc, edst, el, er, emax, denom, E);
  edge_scatter_kernel<8, 16><<<cdiv((long)E * 8, B), B, 0, stream>>>(
      esrc, edst, el, er, emax, denom, bufA, bufB, E);
  finalize_kernel<true>
      <<<cdiv((long)N * 128, B), B, 0, stream>>>(bufB, b1, N, 128);

  // ------------------------- Layer 2: 128 -> 1x64 -------------------------
  gemm_wmma_kernel<128, 64>
      <<<cdiv((long)NSTRIPS * 32, B), B, 0, stream>>>(bufB, W2, bufA, NSTRIPS);
  attn_scores_kernel<1, 64>
      <<<cdiv((long)N, B), B, 0, stream>>>(bufA, al2, ar2, el, er, N);
  fill_kernel<<<cdiv((long)N, B), B, 0, stream>>>(emax, NEG_INF, N);
  hipMemsetAsync(denom, 0, (size_t)N * sizeof(float), stream);
  hipMemsetAsync(outp, 0, (size_t)N * 64 * sizeof(float), stream);
  edge_max_kernel<1><<<cdiv(E, B), B, 0, stream>>>(esrc, edst, el, er, emax, E);
  edge_denom_kernel<1>
      <<<cdiv(E, B), B, 0, stream>>>(esrc, edst, el, er, emax, denom, E);
  edge_scatter_kernel<1, 64><<<cdiv(E, B), B, 0, stream>>>(
      esrc, edst, el, er, emax, denom, bufA, outp, E);
  finalize_kernel<false>
      <<<cdiv((long)N * 64, B), B, 0, stream>>>(outp, b2, N, 64);
}